// LinearRNN_Block_89610197663973
// MI455X (gfx1250) — compile-verified
//
#include <hip/hip_runtime.h>
#include <hip/hip_bf16.h>

#define B_ 8
#define S_ 2048
#define I_ 512
#define H_ 1024

typedef __attribute__((ext_vector_type(2))) float v2f;
typedef __attribute__((ext_vector_type(8))) float v8f;

// D = A(16x4,f32) * B(4x16,f32) + C(16x16,f32)
__device__ __forceinline__ v8f wmma_f32(v2f a, v2f b, v8f c) {
    return __builtin_amdgcn_wmma_f32_16x16x4_f32(false, a, false, b, (short)0, c,
                                                 false, false);
}

// ---------------------------------------------------------------------------
// Kernel 1: P = x @ Wx^T + bx  (into d_out), R = x @ Wr^T + br (into ws)
// Block = 8 waves sharing ONE 64-column weight slice (WGP$ reuse), each wave
// a different 16-row M tile -> block tile is 128(M) x 64(N), K = 512.
// K loop software-pipelined with register double-buffering.
// ---------------------------------------------------------------------------
__global__ void __launch_bounds__(256) proj_kernel(
    const float* __restrict__ x, const float* __restrict__ Wx,
    const float* __restrict__ bx, const float* __restrict__ Wr,
    const float* __restrict__ br, float* __restrict__ P, float* __restrict__ R) {
    const int wave  = threadIdx.x >> 5;
    const int lane  = threadIdx.x & 31;
    const int ncol  = blockIdx.x & 15;                 // 64-col slice, shared
    const int mtile = (blockIdx.x >> 4) * 8 + wave;    // per-wave M tile
    const int m     = lane & 15;
    const int hi    = lane >> 4;

    const float* xrow = x + (mtile * 16 + m) * I_;
    const float* wx0  = Wx + (ncol * 64 + m) * I_;     // + j*16*I_ per tile
    const float* wr0  = Wr + (ncol * 64 + m) * I_;

    v8f accP[4], accR[4];
#pragma unroll
    for (int j = 0; j < 4; ++j) { accP[j] = (v8f)0.0f; accR[j] = (v8f)0.0f; }

    // ---- pipeline prologue: fragments for k = 0
    v2f a_c, bw_c[4], rw_c[4];
    {
        const int kb = 2 * hi;
        a_c = *(const v2f*)(xrow + kb);
#pragma unroll
        for (int j = 0; j < 4; ++j) {
            bw_c[j] = *(const v2f*)(wx0 + j * 16 * I_ + kb);
            rw_c[j] = *(const v2f*)(wr0 + j * 16 * I_ + kb);
        }
    }

#pragma unroll 4
    for (int k = 0; k < I_ - 4; k += 4) {
        const int kb = k + 4 + 2 * hi;                 // next iteration
        v2f a_n = *(const v2f*)(xrow + kb);
        v2f bw_n[4], rw_n[4];
#pragma unroll
        for (int j = 0; j < 4; ++j) {
            bw_n[j] = *(const v2f*)(wx0 + j * 16 * I_ + kb);
            rw_n[j] = *(const v2f*)(wr0 + j * 16 * I_ + kb);
        }
#pragma unroll
        for (int j = 0; j < 4; ++j) {                  // consume current
            accP[j] = wmma_f32(a_c, bw_c[j], accP[j]);
            accR[j] = wmma_f32(a_c, rw_c[j], accR[j]);
        }
        a_c = a_n;
#pragma unroll
        for (int j = 0; j < 4; ++j) { bw_c[j] = bw_n[j]; rw_c[j] = rw_n[j]; }
    }
#pragma unroll
    for (int j = 0; j < 4; ++j) {                      // epilogue k = I_-4
        accP[j] = wmma_f32(a_c, bw_c[j], accP[j]);
        accR[j] = wmma_f32(a_c, rw_c[j], accR[j]);
    }

#pragma unroll
    for (int j = 0; j < 4; ++j) {
        const int ocol = ncol * 64 + j * 16 + m;
        const float bxv = bx[ocol];
        const float brv = br[ocol];
#pragma unroll
        for (int v = 0; v < 8; ++v) {
            const int orow = mtile * 16 + v + 8 * hi;
            P[orow * H_ + ocol] = accP[j][v] + bxv;
            R[orow * H_ + ocol] = accR[j][v] + brv;
        }
    }
}

// ---------------------------------------------------------------------------
// Kernel 2: sequential scan  h_t = xproj_t + h_{t-1} @ Wh^T, in-place in P.
// 64 persistent workgroups; WG g owns output columns [16g, 16g+16).
// 8 waves split K=1024 (128 each). The Wh B-fragments are time-invariant:
// preloaded ONCE into 32 v2f registers per wave. Per step: batch-issue 32
// A-fragment loads (h_{t-1}, L2-hot 64KB buffer), drain 32-deep WMMA chain,
// LDS-reduce the 8 K-partials, add xproj, device-wide barrier.
// hbuf: double-buffered [2][16][H_], rows 8..15 kept zero (M padding).
// ---------------------------------------------------------------------------
__global__ void __launch_bounds__(256) rnn_scan_kernel(
    const float* __restrict__ Wh, float* __restrict__ P,
    float* __restrict__ hbuf, unsigned* __restrict__ bar) {
    __shared__ float part[8 * 256];  // [wave][v][lane]

    const int wg   = blockIdx.x;          // column group
    const int wave = threadIdx.x >> 5;
    const int lane = threadIdx.x & 31;
    const int m    = lane & 15;
    const int hi   = lane >> 4;
    const int cb   = wg * 16;             // first output column
    const int k0   = wave * 128;          // this wave's K range

    // ---- hoist: Wh slice B-fragments, constant over all 2048 steps
    v2f bfrag[32];
    {
        const float* Whrow = Wh + (cb + m) * H_ + k0 + 2 * hi;
#pragma unroll
        for (int i = 0; i < 32; ++i) bfrag[i] = *(const v2f*)(Whrow + i * 4);
    }

    const float* hp0 = hbuf + m * H_ + k0 + 2 * hi;    // per-lane A base

    for (int t = 0; t < S_; ++t) {
        const float* ha = hp0 + ((t & 1) ? 16 * H_ : 0);
        float* hnext    = hbuf + ((t & 1) ? 0 : 16 * H_);

        // batch-issue all A-fragment loads, then drain the WMMA chain
        v2f afrag[32];
#pragma unroll
        for (int i = 0; i < 32; ++i) afrag[i] = *(const v2f*)(ha + i * 4);
        v8f acc = (v8f)0.0f;
#pragma unroll
        for (int i = 0; i < 32; ++i) acc = wmma_f32(afrag[i], bfrag[i], acc);

#pragma unroll
        for (int v = 0; v < 8; ++v) part[wave * 256 + v * 32 + lane] = acc[v];
        __syncthreads();                      // (a) partials visible in LDS

        if (wave == 0) {
#pragma unroll
            for (int v = 0; v < 8; ++v) {
                float sum = 0.f;
#pragma unroll
                for (int w = 0; w < 8; ++w) sum += part[w * 256 + v * 32 + lane];
                const int om = v + 8 * hi;    // output row (batch) 0..15
                float val = 0.f;
                if (om < 8) {                 // real batch rows only
                    const int pidx = (om * S_ + t) * H_ + cb + m;
                    val = sum + P[pidx];      // + xproj_t
                    P[pidx] = val;            // h_seq[t] overwrites xproj[t]
                }
                hnext[om * H_ + cb + m] = val;  // rows 8..15 stay zero
            }
            __threadfence();                  // make h_t visible device-wide
        }
        __syncthreads();                      // (b) wave0 done; part reusable

        if (threadIdx.x == 0) {
            __hip_atomic_fetch_add(bar, 1u, __ATOMIC_RELEASE,
                                   __HIP_MEMORY_SCOPE_AGENT);
            const unsigned target = (unsigned)(t + 1) * gridDim.x;
            while (__hip_atomic_load(bar, __ATOMIC_ACQUIRE,
                                     __HIP_MEMORY_SCOPE_AGENT) < target) {
                __builtin_amdgcn_s_sleep(2);
            }
        }
        __syncthreads();                      // (c) whole block may advance
    }
}

// ---------------------------------------------------------------------------
// Kernel 3: Y = LayerNorm(h_seq + residual) * gamma + beta, in place in d_out.
// One 256-thread block per row of H_=1024 (4 elems/thread), wave32 reductions.
// ---------------------------------------------------------------------------
__global__ void __launch_bounds__(256) ln_kernel(
    float* __restrict__ Y, const float* __restrict__ Rres,
    const float* __restrict__ g, const float* __restrict__ b) {
    __shared__ float red[16];
    const int row  = blockIdx.x;
    const int t    = threadIdx.x;
    const int base = row * H_;

    float vals[4];
    float s = 0.f, s2 = 0.f;
#pragma unroll
    for (int j = 0; j < 4; ++j) {
        const int i = t + j * 256;
        const float v = Y[base + i] + Rres[base + i];
        vals[j] = v;
        s += v;
        s2 += v * v;
    }
#pragma unroll
    for (int off = 16; off > 0; off >>= 1) {
        s  += __shfl_xor(s, off, 32);
        s2 += __shfl_xor(s2, off, 32);
    }
    const int wave = t >> 5, lane = t & 31;
    if (lane == 0) { red[wave] = s; red[8 + wave] = s2; }
    __syncthreads();
    float ts = 0.f, ts2 = 0.f;
#pragma unroll
    for (int w = 0; w < 8; ++w) { ts += red[w]; ts2 += red[8 + w]; }
    const float mu  = ts * (1.0f / H_);
    const float var = ts2 * (1.0f / H_) - mu * mu;
    const float inv = rsqrtf(var + 1e-5f);
#pragma unroll
    for (int j = 0; j < 4; ++j) {
        const int i = t + j * 256;
        Y[base + i] = (vals[j] - mu) * inv * g[i] + b[i];
    }
}

// ---------------------------------------------------------------------------
extern "C" void kernel_launch(void* const* d_in, const int* in_sizes, int n_in,
                              void* d_out, int out_size, void* d_ws, size_t ws_size,
                              hipStream_t stream) {
    const float* x     = (const float*)d_in[0];
    const float* Wx    = (const float*)d_in[1];
    const float* bx    = (const float*)d_in[2];
    const float* Wh    = (const float*)d_in[3];
    const float* Wr    = (const float*)d_in[4];
    const float* br    = (const float*)d_in[5];
    const float* gamma = (const float*)d_in[6];
    const float* beta  = (const float*)d_in[7];

    float* out  = (float*)d_out;                       // xproj -> h_seq -> LN
    float* res  = (float*)d_ws;                        // residual, 64 MB
    float* hbuf = res + (size_t)B_ * S_ * H_;          // [2][16][H_]
    unsigned* bar = (unsigned*)(hbuf + 2 * 16 * H_);   // barrier counter

    // Zero h double-buffer + barrier counter (deterministic per launch).
    hipMemsetAsync(hbuf, 0, (size_t)(2 * 16 * H_) * sizeof(float) + 64, stream);

    // 128 M-blocks x 16 column-slices; 8 waves/block, one M-tile per wave.
    proj_kernel<<<2048, 256, 0, stream>>>(x, Wx, bx, Wr, br, out, res);

    // Persistent scan: 64 WGs (one 16-column slice each), 2048 steps.
    rnn_scan_kernel<<<64, 256, 0, stream>>>(Wh, out, hbuf, bar);

    // Fused residual-add + LayerNorm, one block per (b,s) row.
    ln_kernel<<<B_ * S_, 256, 0, stream>>>(out, res, gamma, beta);
}